// AttentionSheafLearner_81484119540401
// MI455X (gfx1250) — compile-verified
//
#include <hip/hip_runtime.h>

#define C_IN       128
#define YSTRIDE    64      // padded per-node row in workspace: y1 @ col 0..24, y2 @ col 32..56
#define WLDS_PITCH 132     // 128 + 4 pad -> conflict-free ds_load_b64 B-fragment reads

typedef __attribute__((ext_vector_type(2))) float v2f;
typedef __attribute__((ext_vector_type(8))) float v8f;

// ---------------------------------------------------------------------------
// Kernel 1: y[n][0..24] = x[n] @ W1.T ; y[n][32..56] = x[n] @ W2.T
// One wave per 16-node M-tile; 4 WMMA N-tiles (64 cols, padded rows are zero).
// v_wmma_f32_16x16x4_f32, K-loop over 128 in steps of 4.
// ---------------------------------------------------------------------------
__global__ __launch_bounds__(256) void sheaf_node_proj(
    const float* __restrict__ x, const float* __restrict__ W,
    float* __restrict__ y, int N) {
  __shared__ float Wlds[64 * WLDS_PITCH];

  const int tid = threadIdx.x;

  // Stage W into LDS, remapped: rows 0..24 = W[:, :128]; rows 32..56 = W[:, 128:]; rest 0.
  for (int i = tid; i < 64 * C_IN; i += 256) {
    int r = i >> 7;
    int c = i & (C_IN - 1);
    float v = 0.0f;
    if (r < 25)                 v = W[r * 256 + c];
    else if (r >= 32 && r < 57) v = W[(r - 32) * 256 + 128 + c];
    Wlds[r * WLDS_PITCH + c] = v;
  }
  __syncthreads();

  const int wave = tid >> 5;
  const int lane = tid & 31;
  const int m0   = (blockIdx.x * 8 + wave) * 16;
  if (m0 >= N) return;                       // wave-uniform: EXEC stays all-ones for WMMA

  const int idx  = lane & 15;                // M (for A) / N (for B/D)
  const int half = lane >> 4;                // selects K-pair {0,1} vs {2,3}

  const int mrow = (m0 + idx < N) ? (m0 + idx) : (N - 1);
  const float* __restrict__ xrow = x + (size_t)mrow * C_IN;
  const float* __restrict__ wl   = Wlds + idx * WLDS_PITCH;

  v8f acc0 = {}, acc1 = {}, acc2 = {}, acc3 = {};

  for (int k = 0; k < C_IN; k += 4) {
    const int ko = k + 2 * half;             // 8B-aligned (k%4==0, 2*half even)
    v2f a  = *(const v2f*)(xrow + ko);
    v2f b0 = *(const v2f*)(wl + 0 * 16 * WLDS_PITCH + ko);
    v2f b1 = *(const v2f*)(wl + 1 * 16 * WLDS_PITCH + ko);
    v2f b2 = *(const v2f*)(wl + 2 * 16 * WLDS_PITCH + ko);
    v2f b3 = *(const v2f*)(wl + 3 * 16 * WLDS_PITCH + ko);
    acc0 = __builtin_amdgcn_wmma_f32_16x16x4_f32(false, a, false, b0, (short)0, acc0, false, false);
    acc1 = __builtin_amdgcn_wmma_f32_16x16x4_f32(false, a, false, b1, (short)0, acc1, false, false);
    acc2 = __builtin_amdgcn_wmma_f32_16x16x4_f32(false, a, false, b2, (short)0, acc2, false, false);
    acc3 = __builtin_amdgcn_wmma_f32_16x16x4_f32(false, a, false, b3, (short)0, acc3, false, false);
  }

  // D layout: lane<16 -> M=vgpr, lane>=16 -> M=vgpr+8 ; N = idx (+16 per tile)
#pragma unroll
  for (int v = 0; v < 8; ++v) {
    const int mr = m0 + v + 8 * half;
    if (mr < N) {
      float* __restrict__ yr = y + (size_t)mr * YSTRIDE;
      yr[ 0 + idx] = acc0[v];
      yr[16 + idx] = acc1[v];
      yr[32 + idx] = acc2[v];
      yr[48 + idx] = acc3[v];
    }
  }
}

// ---------------------------------------------------------------------------
// Kernel 2: per edge e: maps = y1[row] + y2[col]; out = I - softmax(maps, rows)
// One thread per edge; gathers hit L2 (y is ~10MB << 192MB). Results staged in
// LDS so each wave emits one fully-coalesced 800-float contiguous store.
// ---------------------------------------------------------------------------
__global__ __launch_bounds__(256) void sheaf_edge(
    const int* __restrict__ ei, const float* __restrict__ y,
    float* __restrict__ out, int E) {
  __shared__ float sbuf[256 * 26];           // pitch 26 -> conflict-free LDS

  const int tid = threadIdx.x;
  const int e   = blockIdx.x * 256 + tid;
  const int ec  = (e < E) ? e : (E - 1);

  const int row = ei[ec];
  const int col = ei[E + ec];

  const float* __restrict__ pr = y + (size_t)row * YSTRIDE;       // y1 @ col 0
  const float* __restrict__ pc = y + (size_t)col * YSTRIDE + 32;  // y2 @ col 32

  float m[25];
#pragma unroll
  for (int j = 0; j < 24; j += 4) {          // 16B-aligned float4 gathers
    float4 a = *(const float4*)(pr + j);
    float4 b = *(const float4*)(pc + j);
    m[j + 0] = a.x + b.x;
    m[j + 1] = a.y + b.y;
    m[j + 2] = a.z + b.z;
    m[j + 3] = a.w + b.w;
  }
  m[24] = pr[24] + pc[24];

  float* __restrict__ srow = sbuf + tid * 26;
#pragma unroll
  for (int r = 0; r < 5; ++r) {
    float v0 = m[5 * r + 0], v1 = m[5 * r + 1], v2 = m[5 * r + 2];
    float v3 = m[5 * r + 3], v4 = m[5 * r + 4];
    float mx = fmaxf(fmaxf(fmaxf(v0, v1), fmaxf(v2, v3)), v4);
    float e0 = __expf(v0 - mx), e1 = __expf(v1 - mx), e2 = __expf(v2 - mx);
    float e3 = __expf(v3 - mx), e4 = __expf(v4 - mx);
    float inv = __fdividef(1.0f, e0 + e1 + e2 + e3 + e4);
    srow[5 * r + 0] = (r == 0 ? 1.0f : 0.0f) - e0 * inv;
    srow[5 * r + 1] = (r == 1 ? 1.0f : 0.0f) - e1 * inv;
    srow[5 * r + 2] = (r == 2 ? 1.0f : 0.0f) - e2 * inv;
    srow[5 * r + 3] = (r == 3 ? 1.0f : 0.0f) - e3 * inv;
    srow[5 * r + 4] = (r == 4 ? 1.0f : 0.0f) - e4 * inv;
  }
  __syncthreads();

  // Coalesced writeback: each wave stores its 32 edges * 25 floats contiguously.
  const int wave = tid >> 5;
  const int lane = tid & 31;
  const long long gbase = (long long)(blockIdx.x * 256 + wave * 32) * 25;
  const long long lim   = (long long)E * 25;
  const float* __restrict__ wsrc = sbuf + wave * 32 * 26;
#pragma unroll
  for (int i = lane; i < 800; i += 32) {
    long long gi = gbase + i;
    if (gi < lim) out[gi] = wsrc[(i / 25) * 26 + (i % 25)];
  }
}

// ---------------------------------------------------------------------------
extern "C" void kernel_launch(void* const* d_in, const int* in_sizes, int n_in,
                              void* d_out, int out_size, void* d_ws, size_t ws_size,
                              hipStream_t stream) {
  const float* x  = (const float*)d_in[0];   // [N, 128] fp32
  const float* W  = (const float*)d_in[1];   // [25, 256] fp32
  const int*   ei = (const int*)d_in[2];     // [2, E] int
  float*       out = (float*)d_out;          // [E, 5, 5] fp32
  float*       y   = (float*)d_ws;           // [N, 64] fp32 scratch (~12.8 MB)

  const int N = in_sizes[0] / C_IN;
  const int E = in_sizes[2] / 2;

  const int mtiles  = (N + 15) / 16;
  const int blocks1 = (mtiles + 7) / 8;      // 8 waves (M-tiles) per 256-thread block
  sheaf_node_proj<<<blocks1, 256, 0, stream>>>(x, W, y, N);

  const int blocks2 = (E + 255) / 256;
  sheaf_edge<<<blocks2, 256, 0, stream>>>(ei, y, out, E);
}